// LocalGrouper_withoutKNN_81836306858510
// MI455X (gfx1250) — compile-verified
//
#include <hip/hip_runtime.h>

#define B_   16
#define N_   16384
#define D_   128
#define S_   (N_ / 4)      // 4096
#define TPB  1024          // 32 wave32 waves -> one WGP
#define PPT  (N_ / TPB)    // 16 points per thread (register-resident dist)

#ifndef __has_builtin
#define __has_builtin(x) 0
#endif

#if __has_builtin(__builtin_amdgcn_global_load_async_to_lds_b32)
#define HAVE_ASYNC_LDS 1
typedef __attribute__((address_space(1))) int as1_int;
typedef __attribute__((address_space(3))) int as3_int;
#else
#define HAVE_ASYNC_LDS 0
#endif

// ---------------------------------------------------------------------------
// Kernel 1: FPS per batch (1 workgroup per batch), LDS-resident xyz,
// register-resident running min-distance, 1 barrier per step, plus in-kernel
// bitmask compaction that emits the selected indices already sorted.
// ---------------------------------------------------------------------------
__global__ __launch_bounds__(TPB, 1)
void fps_sorted_kernel(const float* __restrict__ xyz, int* __restrict__ sorted_idx)
{
    extern __shared__ float smem[];
    float*    lx    = smem;                 // [N]
    float*    ly    = lx + N_;              // [N]
    float*    lz    = ly + N_;              // [N]
    float*    redv  = lz + N_;              // [2][32] ping-pong wave maxima
    int*      redi  = (int*)(redv + 64);    // [2][32]
    unsigned* lmask = (unsigned*)(redi + 64);   // [N/32] selection bitmask
    int*      lbase = (int*)(lmask + (N_ / 32)); // [N/32] exclusive popcount scan

    const int t = threadIdx.x;
    const int b = blockIdx.x;
    const float* gx = xyz + (size_t)b * N_ * 3;

    // ---- fill LDS with this batch's xyz (SoA) via async global->LDS DMA ----
#if HAVE_ASYNC_LDS
    #pragma unroll
    for (int k = 0; k < PPT; ++k) {
        const int n = t + k * TPB;
        const float* src = gx + n * 3;
        __builtin_amdgcn_global_load_async_to_lds_b32(
            (as1_int*)(src + 0), (as3_int*)&lx[n], 0, 0);
        __builtin_amdgcn_global_load_async_to_lds_b32(
            (as1_int*)(src + 1), (as3_int*)&ly[n], 0, 0);
        __builtin_amdgcn_global_load_async_to_lds_b32(
            (as1_int*)(src + 2), (as3_int*)&lz[n], 0, 0);
    }
#if __has_builtin(__builtin_amdgcn_s_wait_asynccnt)
    __builtin_amdgcn_s_wait_asynccnt(0);
#else
    asm volatile("s_wait_asynccnt 0" ::: "memory");
#endif
#else
    #pragma unroll
    for (int k = 0; k < PPT; ++k) {
        const int n = t + k * TPB;
        const float* src = gx + n * 3;
        lx[n] = src[0]; ly[n] = src[1]; lz[n] = src[2];
    }
#endif

    // mask init; index 0 is always the first FPS pick (reference starts at 0)
    if (t < N_ / 32) lmask[t] = (t == 0) ? 1u : 0u;
    __syncthreads();

    // register copies + running min-distance
    float px[PPT], py[PPT], pz[PPT], dmin[PPT];
    #pragma unroll
    for (int k = 0; k < PPT; ++k) {
        const int n = t + k * TPB;
        px[k] = lx[n]; py[k] = ly[n]; pz[k] = lz[n];
        dmin[k] = 1e10f;   // BIG from reference
    }

    int far = 0;
    for (int s = 1; s < S_; ++s) {
        // centroid broadcast-read from LDS (uniform address across lanes)
        const float cx = lx[far], cy = ly[far], cz = lz[far];

        float lv = -1.0f;
        int   li = 0;
        #pragma unroll
        for (int k = 0; k < PPT; ++k) {
            // exact order of reference: (dx*dx + dy*dy) + dz*dz, no FMA contraction
            const float dx = __fsub_rn(px[k], cx);
            const float dy = __fsub_rn(py[k], cy);
            const float dz = __fsub_rn(pz[k], cz);
            const float d  = __fadd_rn(__fadd_rn(__fmul_rn(dx, dx), __fmul_rn(dy, dy)),
                                       __fmul_rn(dz, dz));
            const float nd = fminf(dmin[k], d);
            dmin[k] = nd;
            // strict '>' keeps lowest index on ties (n ascends with k)
            if (nd > lv) { lv = nd; li = t + k * TPB; }
        }

        // wave32 argmax (ties -> lowest index, matching jnp.argmax)
        #pragma unroll
        for (int off = 16; off > 0; off >>= 1) {
            const float ov = __shfl_xor(lv, off, 32);
            const int   oi = __shfl_xor(li, off, 32);
            if (ov > lv || (ov == lv && oi < li)) { lv = ov; li = oi; }
        }

        const int wave = t >> 5;
        const int ph   = (s & 1) * 32;        // ping-pong: 1 barrier per step
        if ((t & 31) == 0) { redv[ph + wave] = lv; redi[ph + wave] = li; }
        __syncthreads();

        // cross-wave argmax: lane w loads wave w's result, shuffle-reduce again
        float bv = redv[ph + (t & 31)];
        int   bi = redi[ph + (t & 31)];
        #pragma unroll
        for (int off = 16; off > 0; off >>= 1) {
            const float ov = __shfl_xor(bv, off, 32);
            const int   oi = __shfl_xor(bi, off, 32);
            if (ov > bv || (ov == bv && oi < bi)) { bv = ov; bi = oi; }
        }
        far = bi;                              // uniform across the block
        if (t == 0) lmask[far >> 5] |= (1u << (far & 31));
    }
    __syncthreads();

    // ---- compaction: emit selected indices in ascending order (== sort) ----
    if (t == 0) {
        int run = 0;
        for (int w = 0; w < N_ / 32; ++w) { lbase[w] = run; run += __popc(lmask[w]); }
    }
    __syncthreads();
    if (t < N_ / 32) {
        unsigned m   = lmask[t];
        int      pos = lbase[t];
        int* outp = sorted_idx + b * S_;
        while (m) {
            const int bit = __builtin_ctz(m);
            outp[pos++] = (t << 5) + bit;
            m &= m - 1;
        }
    }
}

// ---------------------------------------------------------------------------
// Kernel 2: gather + concat. One wave per output row; float4 reads.
// d_out layout: new_xyz [B,S,3] | new_points [B,S,1,131] | points_res [B,S,128]
// ---------------------------------------------------------------------------
__global__ __launch_bounds__(256)
void gather_concat_kernel(const float* __restrict__ xyz,
                          const float* __restrict__ points,
                          const float* __restrict__ points_res,
                          const int*   __restrict__ sorted_idx,
                          float* __restrict__ out)
{
    const int lane = threadIdx.x & 31;
    const int wave = threadIdx.x >> 5;
    const int row  = blockIdx.x * 8 + wave;     // row = b*S + s, in [0, B*S)
    const int b    = row / S_;
    const int n    = sorted_idx[row];

    float* out_xyz = out;                                   // B*S*3
    float* out_np  = out + (size_t)B_ * S_ * 3;             // B*S*(D+3)
    float* out_res = out_np + (size_t)B_ * S_ * (D_ + 3);   // B*S*D

    const float* xrow = xyz        + ((size_t)b * N_ + n) * 3;
    const float* prow = points     + ((size_t)b * N_ + n) * D_;
    const float* rrow = points_res + ((size_t)b * N_ + n) * D_;

    float* xyz_row = out_xyz + (size_t)row * 3;
    float* np_row  = out_np  + (size_t)row * (D_ + 3);
    float* res_row = out_res + (size_t)row * D_;

    if (lane < 3) {
        const float v = xrow[lane];
        xyz_row[lane] = v;       // new_xyz
        np_row[lane]  = v;       // concat prefix
    }

    // 32 lanes x float4 == 128 floats per row
    const float4 p = ((const float4*)prow)[lane];
    np_row[3 + lane * 4 + 0] = p.x;
    np_row[3 + lane * 4 + 1] = p.y;
    np_row[3 + lane * 4 + 2] = p.z;
    np_row[3 + lane * 4 + 3] = p.w;

    const float4 r = ((const float4*)rrow)[lane];
    ((float4*)res_row)[lane] = r;
}

// ---------------------------------------------------------------------------
extern "C" void kernel_launch(void* const* d_in, const int* in_sizes, int n_in,
                              void* d_out, int out_size, void* d_ws, size_t ws_size,
                              hipStream_t stream)
{
    (void)in_sizes; (void)n_in; (void)out_size; (void)ws_size;

    const float* xyz        = (const float*)d_in[0];
    const float* points     = (const float*)d_in[1];
    const float* points_res = (const float*)d_in[2];
    float* out    = (float*)d_out;
    int*   sorted = (int*)d_ws;   // B*S ints = 256 KB scratch

    // LDS: 3*N xyz + 2*32 (val+idx) ping-pong + N/32 mask + N/32 scan
    const size_t lds_bytes =
        (size_t)(3 * N_ + 64 + 64 + (N_ / 32) + (N_ / 32)) * sizeof(float); // 201216 B

    // opt in to >64KB dynamic LDS (WGP has 320KB); capture-safe (not a stream op)
    (void)hipFuncSetAttribute((const void*)fps_sorted_kernel,
                              hipFuncAttributeMaxDynamicSharedMemorySize,
                              (int)lds_bytes);

    fps_sorted_kernel<<<B_, TPB, lds_bytes, stream>>>(xyz, sorted);
    gather_concat_kernel<<<(B_ * S_) / 8, 256, 0, stream>>>(xyz, points, points_res,
                                                            sorted, out);
}